// PoseToKeypoints_66331474919799
// MI455X (gfx1250) — compile-verified
//
#include <hip/hip_runtime.h>

typedef __attribute__((ext_vector_type(2))) float v2f;
typedef __attribute__((ext_vector_type(4))) float v4f;
typedef __attribute__((ext_vector_type(8))) float v8f;

#define NPTS 128

// Fast reciprocal: v_rcp_f32 + one Newton-Raphson step (~1 ulp).
__device__ __forceinline__ float fast_rcp(float x) {
  float r = __builtin_amdgcn_rcpf(x);
  float e = __builtin_fmaf(-x, r, 1.0f);
  return __builtin_fmaf(r, e, r);
}

// One wave handles 16 poses x 128 keypoints.
// Per 16-point block: 3x v_wmma_f32_16x16x4_f32 (rows i=0,1,2 of the
// projected homogeneous coordinates), then perspective divide + scale.
__global__ __launch_bounds__(256) void pose_kp_wmma(
    const float* __restrict__ quat,   // (B,4)
    const float* __restrict__ kp,     // (3,128)
    const float* __restrict__ cm,     // (3,3)
    float* __restrict__ out)          // (B,2,128)
{
  const int lane = threadIdx.x & 31;
  const int wave = threadIdx.x >> 5;
  const int half = lane >> 4;   // 0: K=0..1 slice, 1: K=2..3 slice
  const int col  = lane & 15;
  const int b0   = (blockIdx.x * 8 + wave) * 16;   // first pose of this wave's tile

  // ---- quaternion -> R (lane m = col owns pose b0+col; upper half duplicates) ----
  const v4f q = *(const v4f*)(quat + (size_t)(b0 + col) * 4);
  const float x = q.x, y = q.y, z = q.z, w = q.w;
  const float xx = x*x, yy = y*y, zz = z*z;
  const float xy = x*y, xz = x*z, yz = y*z;
  const float xw = x*w, yw = y*w, zw = z*w;
  const float r00 = 1.f - 2.f*(yy+zz), r01 = 2.f*(xy-zw),       r02 = 2.f*(xz+yw);
  const float r10 = 2.f*(xy+zw),       r11 = 1.f - 2.f*(xx+zz), r12 = 2.f*(yz-xw);
  const float r20 = 2.f*(xz-yw),       r21 = 2.f*(yz+xw),       r22 = 1.f - 2.f*(xx+yy);

  // camera matrix: wave-uniform -> scalar loads
  const float c00 = cm[0], c01 = cm[1], c02 = cm[2];
  const float c10 = cm[3], c11 = cm[4], c12 = cm[5];
  const float c20 = cm[6], c21 = cm[7], c22 = cm[8];

  // M = C * R ; t' = C * [0,0,0.7]
  const float m00 = c00*r00 + c01*r10 + c02*r20;
  const float m01 = c00*r01 + c01*r11 + c02*r21;
  const float m02 = c00*r02 + c01*r12 + c02*r22;
  const float m10 = c10*r00 + c11*r10 + c12*r20;
  const float m11 = c10*r01 + c11*r11 + c12*r21;
  const float m12 = c10*r02 + c11*r12 + c12*r22;
  const float m20 = c20*r00 + c21*r10 + c22*r20;
  const float m21 = c20*r01 + c21*r11 + c22*r21;
  const float m22 = c20*r02 + c21*r12 + c22*r22;
  const float tp0 = 0.7f * c02, tp1 = 0.7f * c12, tp2 = 0.7f * c22;

  // A-matrix lanes (16x4 f32 layout): lanes 0-15 = {K0,K1}, lanes 16-31 = {K2,K3}
  v2f a0, a1, a2;
  a0.x = half ? m02 : m00;  a0.y = half ? tp0 : m01;
  a1.x = half ? m12 : m10;  a1.y = half ? tp1 : m11;
  a2.x = half ? m22 : m20;  a2.y = half ? tp2 : m21;

  // B-matrix (4x16 f32 layout): lanes 0-15 = {kp0[n],kp1[n]}, lanes 16-31 = {kp2[n],1}
  v2f bmat[8];
  const int krow0 = half ? 2 : 0;
#pragma unroll
  for (int pb = 0; pb < 8; ++pb) {
    const int n = pb * 16 + col;
    const float blo = kp[krow0 * NPTS + n];
    const float bh  = kp[1 * NPTS + n];
    v2f bv; bv.x = blo; bv.y = half ? 1.0f : bh;
    bmat[pb] = bv;
  }

  const float sx = 1.0f / 640.0f, sy = 1.0f / 480.0f;
  const v8f zero8 = {};

#pragma unroll
  for (int pb = 0; pb < 8; ++pb) {
    // (neg_a, A, neg_b, B, c_mod, C, reuse_a, reuse_b)
    v8f d0 = __builtin_amdgcn_wmma_f32_16x16x4_f32(
        false, a0, false, bmat[pb], (short)0, zero8, false, false);
    v8f d1 = __builtin_amdgcn_wmma_f32_16x16x4_f32(
        false, a1, false, bmat[pb], (short)0, zero8, false, false);
    v8f d2 = __builtin_amdgcn_wmma_f32_16x16x4_f32(
        false, a2, false, bmat[pb], (short)0, zero8, false, false);

    const int n = pb * 16 + col;
#pragma unroll
    for (int v = 0; v < 8; ++v) {
      const int brow = b0 + v + 8 * half;            // pose index for this D row
      const float inv = fast_rcp(d2[v]);
      const float o0 = d0[v] * inv * sx;
      const float o1 = d1[v] * inv * sy;
      float* p = out + (size_t)brow * (2 * NPTS) + n;
      __builtin_nontemporal_store(o0, p);            // channel 0
      __builtin_nontemporal_store(o1, p + NPTS);     // channel 1
    }
  }
}

extern "C" void kernel_launch(void* const* d_in, const int* in_sizes, int n_in,
                              void* d_out, int out_size, void* d_ws, size_t ws_size,
                              hipStream_t stream) {
  const float* quat = (const float*)d_in[0];   // (B,4) f32
  const float* kp   = (const float*)d_in[1];   // (3,128) f32
  const float* cm   = (const float*)d_in[2];   // (3,3) f32
  float* out = (float*)d_out;                  // (B,2,128) f32

  const int B = in_sizes[0] / 4;               // 65536
  const int blocks = B / (16 * 8);             // 8 waves/block, 16 poses/wave
  pose_kp_wmma<<<blocks, 256, 0, stream>>>(quat, kp, cm, out);
}